// FaceModel_88587995447602
// MI455X (gfx1250) — compile-verified
//
#include <hip/hip_runtime.h>
#include <math.h>

// ---------------------------------------------------------------------------
// FaceModel GNN pipeline for gfx1250 (MI455X), fp32 end-to-end.
// Dense GEMMs: V_WMMA_F32_16X16X4_F32 fed from LDS; X-tile staged with
// GLOBAL_LOAD_ASYNC_TO_LDS_B128 (ASYNCcnt) when no input activation is fused.
// ---------------------------------------------------------------------------

typedef float v2f __attribute__((ext_vector_type(2)));
typedef float v8f __attribute__((ext_vector_type(8)));

#define FDIM 128
#define XS_STRIDE 36    // floats: 32 + 4 pad; row stride 36 -> conflict-free col reads
#define WS2_STRIDE 144  // float2: 128 + 16 pad; +half-row = 288 floats = +32 banks

__device__ __forceinline__ float eluf(float x) {
    return x > 0.0f ? x : expm1f(x);
}

// Async global->LDS 16B copy (gfx1250 VGLOBAL op 98, tracked by ASYNCcnt).
__device__ __forceinline__ void async_b128_to_lds(unsigned lds_byte_off, const void* gptr) {
    asm volatile("global_load_async_to_lds_b128 %0, %1, off"
                 :: "v"(lds_byte_off), "v"(gptr)
                 : "memory");
}
__device__ __forceinline__ void wait_asynccnt0() {
    asm volatile("s_wait_asynccnt 0x0" ::: "memory");
}

// ---------------------------------------------------------------------------
// Entry MLP layer 0:  out[n][f] = elu( sum_k hv[n][k] * w0[k][f] + b0[f] ), K=7
// ---------------------------------------------------------------------------
__global__ __launch_bounds__(128) void mlp_face0_kernel(
    const float* __restrict__ hv, const float* __restrict__ w0,
    const float* __restrict__ b0, float* __restrict__ out, int N)
{
    const int n = blockIdx.x;
    const int f = threadIdx.x;
    if (n >= N) return;
    float acc = b0[f];
    const float* h = hv + (long long)n * 7;
#pragma unroll
    for (int k = 0; k < 7; ++k)
        acc = fmaf(h[k], w0[k * FDIM + f], acc);
    out[(long long)n * FDIM + f] = eluf(acc);
}

// ---------------------------------------------------------------------------
// Fused WMMA GEMM:  Y[N,128] = post( pre(X)[N,128] @ W[128,128] (+bias) (+Y) )
// Block = 256 threads = 8 wave32; block tile = 128 rows x 128 cols.
// K chunked by 32: Xs[128][36] f32, Ws2[16][144] f32x2 (pair-interleaved so a
// B fragment is one ds_load_b64). 256 v_wmma_f32_16x16x4_f32 per wave.
// ---------------------------------------------------------------------------
__global__ __launch_bounds__(256) void gemm_wmma_kernel(
    const float* __restrict__ X, const float* __restrict__ W,
    const float* __restrict__ bias, float* __restrict__ Y,
    int N, int in_elu, int out_elu, int accum)
{
    __shared__ float Xs[128 * XS_STRIDE];
    __shared__ v2f   Ws2[16 * WS2_STRIDE];

    const int tid  = threadIdx.x;
    const int lane = tid & 31;          // wave32
    const int wave = tid >> 5;          // 0..7
    const int half = lane >> 4;         // 0: K+{0,1}, 1: K+{2,3}
    const int lrow = lane & 15;
    const long long gr0 = (long long)blockIdx.x * 128;

    v8f acc[8] = {};

    for (int kc = 0; kc < 4; ++kc) {
        const int k0 = kc * 32;

        // ---- stage X[gr0..gr0+127][k0..k0+31] ----
        if (!in_elu) {
            // Pure copy: async global->LDS, no VGPR round-trip.
#pragma unroll
            for (int it = 0; it < 4; ++it) {
                const int idx = it * 256 + tid;   // 0..1023
                const int row = idx >> 3;         // 0..127
                const int q   = idx & 7;          // 16B group in 32-col chunk
                const long long grow = gr0 + row;
                if (grow < N) {
                    const unsigned dst = (unsigned)(uintptr_t)(const void*)&Xs[row * XS_STRIDE + q * 4];
                    async_b128_to_lds(dst, (const void*)(X + grow * FDIM + k0 + q * 4));
                }
                // OOB rows: stale LDS only feeds WMMA rows that are masked at store.
            }
        } else {
            // Fused ELU on input: stage through registers.
#pragma unroll
            for (int it = 0; it < 4; ++it) {
                const int idx = it * 256 + tid;
                const int row = idx >> 3;
                const int q   = idx & 7;
                float4 v = make_float4(0.f, 0.f, 0.f, 0.f);
                const long long grow = gr0 + row;
                if (grow < N)
                    v = *(const float4*)(X + grow * FDIM + k0 + q * 4);
                v.x = eluf(v.x); v.y = eluf(v.y);
                v.z = eluf(v.z); v.w = eluf(v.w);
                float* p = &Xs[row * XS_STRIDE + q * 4];
                p[0] = v.x; p[1] = v.y; p[2] = v.z; p[3] = v.w;
            }
        }

        // ---- stage W[k0..k0+31][0..127] pair-interleaved: Ws2[r][c]=(W[2r][c],W[2r+1][c])
#pragma unroll
        for (int it = 0; it < 8; ++it) {
            const int idx = it * 256 + tid;  // 0..2047
            const int r   = idx >> 7;        // k-pair 0..15
            const int c   = idx & 127;       // col
            const long long kr = (long long)(k0 + 2 * r) * FDIM + c;
            v2f w;
            w.x = W[kr];
            w.y = W[kr + FDIM];
            Ws2[r * WS2_STRIDE + c] = w;
        }

        if (!in_elu) wait_asynccnt0();
        __syncthreads();

        // ---- WMMA over this 32-deep K chunk ----
        const float* xrow = &Xs[(wave * 16 + lrow) * XS_STRIDE];
#pragma unroll
        for (int ks = 0; ks < 8; ++ks) {
            const int kk = ks * 4 + 2 * half;                 // lane-half K pair
            const v2f a = *(const v2f*)(xrow + kk);           // one ds_load_b64
            const v2f* brow = &Ws2[(2 * ks + half) * WS2_STRIDE + lrow];
#pragma unroll
            for (int t = 0; t < 8; ++t) {
                const v2f b = brow[t * 16];                   // one ds_load_b64
                acc[t] = __builtin_amdgcn_wmma_f32_16x16x4_f32(
                    /*neg_a=*/false, a, /*neg_b=*/false, b,
                    /*c_mod=*/(short)0, acc[t],
                    /*reuse_a=*/false, /*reuse_b=*/false);
            }
        }
        __syncthreads();
    }

    // ---- epilogue: C/D layout -> VGPR j holds row j (lanes 0-15) / 8+j ----
#pragma unroll
    for (int t = 0; t < 8; ++t) {
        const int col = t * 16 + lrow;
        const float bv = bias ? bias[col] : 0.0f;
#pragma unroll
        for (int j = 0; j < 8; ++j) {
            const long long grow = gr0 + wave * 16 + half * 8 + j;
            if (grow < N) {
                float val = acc[t][j] + bv;
                float* py = Y + grow * FDIM + col;
                if (accum)   val += *py;
                if (out_elu) val  = eluf(val);
                *py = val;
            }
        }
    }
}

// ---------------------------------------------------------------------------
// Zero-fill (float4 grid-stride).
// ---------------------------------------------------------------------------
__global__ __launch_bounds__(256) void zero_f4_kernel(float4* __restrict__ p, long long n4)
{
    long long i = (long long)blockIdx.x * 256 + threadIdx.x;
    const long long stride = (long long)gridDim.x * 256;
    const float4 z = make_float4(0.f, 0.f, 0.f, 0.f);
    for (; i < n4; i += stride) p[i] = z;
}

// ---------------------------------------------------------------------------
// Scatter-add:  agg[dst[e]][:] += x[src[e]][:]
// 32 lanes per edge; float4 gather + 4x global_atomic_add_f32.
// ---------------------------------------------------------------------------
__global__ __launch_bounds__(256) void scatter_add_kernel(
    const float* __restrict__ x, const int* __restrict__ src,
    const int* __restrict__ dst, float* __restrict__ agg, long long E)
{
    const long long g = (long long)blockIdx.x * 256 + threadIdx.x;
    const long long e = g >> 5;
    if (e >= E) return;
    const int c = (int)(g & 31) * 4;
    const int s = src[e];
    const int d = dst[e];
    const float4 v = *(const float4*)(x + (long long)s * FDIM + c);
    float* p = agg + (long long)d * FDIM + c;
    unsafeAtomicAdd(p + 0, v.x);
    unsafeAtomicAdd(p + 1, v.y);
    unsafeAtomicAdd(p + 2, v.z);
    unsafeAtomicAdd(p + 3, v.w);
}

// ---------------------------------------------------------------------------
// Head:  out[n][0..1] = v[n][:] @ w2[128][2] + b2[0..1]
// ---------------------------------------------------------------------------
__global__ __launch_bounds__(256) void head_kernel(
    const float* __restrict__ v, const float* __restrict__ w2,
    const float* __restrict__ b2, float* __restrict__ out, int N)
{
    const int n = blockIdx.x * blockDim.x + threadIdx.x;
    if (n >= N) return;
    float a0 = b2[0], a1 = b2[1];
    const float* row = v + (long long)n * FDIM;
#pragma unroll 4
    for (int k = 0; k < FDIM; ++k) {
        const float x = row[k];
        a0 = fmaf(x, w2[2 * k + 0], a0);
        a1 = fmaf(x, w2[2 * k + 1], a1);
    }
    out[2 * (long long)n + 0] = a0;
    out[2 * (long long)n + 1] = a1;
}

// ---------------------------------------------------------------------------
extern "C" void kernel_launch(void* const* d_in, const int* in_sizes, int n_in,
                              void* d_out, int out_size, void* d_ws, size_t ws_size,
                              hipStream_t stream)
{
    const float* hv        = (const float*)d_in[0];
    const int*   ei        = (const int*)  d_in[1];
    const float* mf_w0     = (const float*)d_in[2];
    const float* mf_b0     = (const float*)d_in[3];
    const float* mf_w1     = (const float*)d_in[4];
    const float* mf_b1     = (const float*)d_in[5];
    const float* g1_rel_w  = (const float*)d_in[6];
    const float* g1_rel_b  = (const float*)d_in[7];
    const float* g1_root_w = (const float*)d_in[8];
    const float* g2_rel_w  = (const float*)d_in[9];
    const float* g2_rel_b  = (const float*)d_in[10];
    const float* g2_root_w = (const float*)d_in[11];
    const float* s1_w0     = (const float*)d_in[12];
    const float* s1_b0     = (const float*)d_in[13];
    const float* s1_w1     = (const float*)d_in[14];
    const float* s1_b1     = (const float*)d_in[15];
    const float* s1_w2     = (const float*)d_in[16];
    const float* s1_b2     = (const float*)d_in[17];
    const float* s2_w0     = (const float*)d_in[18];
    const float* s2_b0     = (const float*)d_in[19];
    const float* s2_w1     = (const float*)d_in[20];
    const float* s2_b1     = (const float*)d_in[21];
    const float* s2_w2     = (const float*)d_in[22];
    const float* s2_b2     = (const float*)d_in[23];

    const int       N = in_sizes[0] / 7;
    const long long E = in_sizes[1] / 2;
    const int* src = ei;
    const int* dst = ei + E;

    // Workspace: 4 x [N,128] f32 ping-pong buffers.
    const size_t bufElems = (size_t)N * FDIM;
    float* A = (float*)d_ws;        // face_features (persists)
    float* B = A + bufElems;
    float* C = B + bufElems;
    float* D = C + bufElems;

    const int gemmGrid = (N + 127) / 128;
    const int sgrid    = (int)((E * 32 + 255) / 256);
    const long long n4 = (long long)N * (FDIM / 4);

    // 1) entry MLP:  t0 = elu(hv @ mf_w0 + mf_b0)           -> B
    mlp_face0_kernel<<<N, 128, 0, stream>>>(hv, mf_w0, mf_b0, B, N);
    // 2) ff = elu(t0 @ mf_w1 + mf_b1)                        -> A
    gemm_wmma_kernel<<<gemmGrid, 256, 0, stream>>>(B, mf_w1, mf_b1, A, N, 0, 1, 0);

    // 3) agg1 = scatter_add(ff)                              -> C
    zero_f4_kernel<<<4096, 256, 0, stream>>>((float4*)C, n4);
    scatter_add_kernel<<<sgrid, 256, 0, stream>>>(A, src, dst, C, E);

    // 4) h = agg1 @ g1_rel_w + g1_rel_b + ff @ g1_root_w     -> B
    gemm_wmma_kernel<<<gemmGrid, 256, 0, stream>>>(C, g1_rel_w, g1_rel_b, B, N, 0, 0, 0);
    gemm_wmma_kernel<<<gemmGrid, 256, 0, stream>>>(A, g1_root_w, nullptr,  B, N, 0, 0, 1);

    // 5) stage1: h2 = elu(elu(elu(elu(h)@w0+b0)@w1+b1)@w2+b2)
    gemm_wmma_kernel<<<gemmGrid, 256, 0, stream>>>(B, s1_w0, s1_b0, D, N, 1, 1, 0); // t1 -> D
    gemm_wmma_kernel<<<gemmGrid, 256, 0, stream>>>(D, s1_w1, s1_b1, C, N, 0, 1, 0); // t2 -> C
    gemm_wmma_kernel<<<gemmGrid, 256, 0, stream>>>(C, s1_w2, s1_b2, B, N, 0, 1, 0); // h2 -> B

    // 6) agg2 = scatter_add(h2)                              -> C
    zero_f4_kernel<<<4096, 256, 0, stream>>>((float4*)C, n4);
    scatter_add_kernel<<<sgrid, 256, 0, stream>>>(B, src, dst, C, E);

    // 7) g = agg2 @ g2_rel_w + g2_rel_b + h2 @ g2_root_w     -> D
    gemm_wmma_kernel<<<gemmGrid, 256, 0, stream>>>(C, g2_rel_w, g2_rel_b, D, N, 0, 0, 0);
    gemm_wmma_kernel<<<gemmGrid, 256, 0, stream>>>(B, g2_root_w, nullptr,  D, N, 0, 0, 1);

    // 8) stage2 on concat [ff, g]: u = ff@W0_top + b0 + g@W0_bot  -> C
    gemm_wmma_kernel<<<gemmGrid, 256, 0, stream>>>(A, s2_w0,               s2_b0,   C, N, 0, 0, 0);
    gemm_wmma_kernel<<<gemmGrid, 256, 0, stream>>>(D, s2_w0 + FDIM * FDIM, nullptr, C, N, 0, 0, 1);
    //    v = elu(elu(u) @ s2_w1 + s2_b1)                     -> B
    gemm_wmma_kernel<<<gemmGrid, 256, 0, stream>>>(C, s2_w1, s2_b1, B, N, 1, 1, 0);

    // 9) out = v @ s2_w2 + s2_b2                             -> d_out [N,2]
    head_kernel<<<(N + 255) / 256, 256, 0, stream>>>(B, s2_w2, s2_b2, (float*)d_out, N);
}